// GraphGATClassifier_6820408066565
// MI455X (gfx1250) — compile-verified
//
#include <hip/hip_runtime.h>
#include <hip/hip_bf16.h>

// ---------------------------------------------------------------------------
// GraphGATClassifier for MI455X (gfx1250, wave32).
// GEMMs use V_WMMA_F32_16X16X4_F32 (exact fp32; workload is HBM-bound on the
// edge gather/scatter so fp32 matrix path is the right precision choice).
// ---------------------------------------------------------------------------

typedef __attribute__((ext_vector_type(2))) float v2f;
typedef __attribute__((ext_vector_type(8))) float v8f;

#define HEADS 4
#define HID   64
#define OUTC  256   // HEADS*HID
#define INC   128
#define NEG_SLOPE 0.2f

// ---------------- utility ----------------

__global__ void fill_f32(float* __restrict__ p, long long n, float v) {
  long long i = (long long)blockIdx.x * blockDim.x + threadIdx.x;
  if (i < n) p[i] = v;
}

__device__ __forceinline__ void atomic_max_f32(float* addr, float val) {
  // sign-split trick: works for mixed-sign floats when init is very negative
  if (val >= 0.0f)
    atomicMax((int*)addr, __float_as_int(val));
  else
    atomicMin((unsigned int*)addr, __float_as_uint(val));
}

// ---------------- WMMA fp32 GEMM: C[rows,256] = A[rows,K] @ W[K,256] --------
// One 16x16 output tile per wave; 8 waves per 256-thread block.
// A layout (16x4 f32): lanes 0-15 -> K=kb+0,kb+1 ; lanes 16-31 -> K=kb+2,kb+3
// B layout (4x16 f32): mirrored; C/D: VGPR r -> M = r + (lane<16 ? 0 : 8).
template<int K>
__global__ void gemm_wmma_f32(const float* __restrict__ A,
                              const float* __restrict__ W,
                              float* __restrict__ C, int nrows) {
  (void)nrows;  // nrows is a multiple of 16 for this problem (50000 = 3125*16)
  const int lane = threadIdx.x & 31;
  const int wave = threadIdx.x >> 5;
  const int tM = blockIdx.x;
  const int tN = (blockIdx.y << 3) + wave;      // 16 column tiles total
  const int mrow = (tM << 4) + (lane & 15);
  const int ncol = (tN << 4) + (lane & 15);
  const int khi  = (lane >> 4) << 1;            // 0 (lanes 0-15) or 2 (16-31)
  const float* __restrict__ arow = A + (long long)mrow * K;

  v8f acc = {};
  for (int kb = 0; kb < K; kb += 4) {
    v2f a, b;
    a.x = arow[kb + khi];
    a.y = arow[kb + khi + 1];
    b.x = W[(long long)(kb + khi)     * OUTC + ncol];
    b.y = W[(long long)(kb + khi + 1) * OUTC + ncol];
    acc = __builtin_amdgcn_wmma_f32_16x16x4_f32(
        /*neg_a=*/false, a, /*neg_b=*/false, b,
        /*c_mod=*/(short)0, acc, /*reuse_a=*/false, /*reuse_b=*/false);
  }

  const int mbase = (tM << 4) + ((lane >> 4) << 3);
#pragma unroll
  for (int r = 0; r < 8; ++r)
    C[(long long)(mbase + r) * OUTC + ncol] = acc[r];
}

// ---------------- attention scores: alpha_{src,dst}[n,h] --------------------
__global__ void attn_scores(const float* __restrict__ H,
                            const float* __restrict__ a_s,
                            const float* __restrict__ a_d,
                            float* __restrict__ AS, float* __restrict__ AD,
                            long long n_nodes) {
  long long idx = (long long)blockIdx.x * blockDim.x + threadIdx.x;  // n*4+h
  if (idx >= n_nodes * HEADS) return;
  long long n = idx >> 2;
  int h = (int)(idx & 3);
  const float* hp = H + n * OUTC + h * HID;
  float ss = 0.f, sd = 0.f;
#pragma unroll 8
  for (int c = 0; c < HID; ++c) {
    float v = hp[c];
    ss += v * a_s[h * HID + c];
    sd += v * a_d[h * HID + c];
  }
  AS[idx] = ss;
  AD[idx] = sd;
}

// ---------------- edge passes ----------------

__device__ __forceinline__ void edge_sd(const int* __restrict__ ei,
                                        long long E, long long e,
                                        int& s, int& d) {
  if (e < E) { s = ei[e]; d = ei[E + e]; }
  else       { s = d = (int)(e - E); }        // self loop
}

// e = leaky_relu(as[src]+ad[dst]); segment max over dst
__global__ void edge_score_max(const float* __restrict__ AS,
                               const float* __restrict__ AD,
                               const int* __restrict__ ei,
                               long long E, long long Etot,
                               float* __restrict__ EW, float* __restrict__ M) {
  long long idx = (long long)blockIdx.x * blockDim.x + threadIdx.x;  // e*4+h
  if (idx >= Etot * HEADS) return;
  long long e = idx >> 2;
  int h = (int)(idx & 3);
  int s, d;
  edge_sd(ei, E, e, s, d);
  float v = AS[(long long)s * HEADS + h] + AD[(long long)d * HEADS + h];
  v = (v > 0.f) ? v : NEG_SLOPE * v;
  EW[idx] = v;
  atomic_max_f32(&M[(long long)d * HEADS + h], v);
}

// ex = exp(e - m[dst]); segment sum over dst; EW overwritten with ex
__global__ void edge_exp_sum(const int* __restrict__ ei,
                             long long E, long long Etot,
                             const float* __restrict__ M,
                             float* __restrict__ EW, float* __restrict__ DEN) {
  long long idx = (long long)blockIdx.x * blockDim.x + threadIdx.x;
  if (idx >= Etot * HEADS) return;
  long long e = idx >> 2;
  int h = (int)(idx & 3);
  int s, d;
  edge_sd(ei, E, e, s, d);
  float ex = __expf(EW[idx] - M[(long long)d * HEADS + h]);
  EW[idx] = ex;
  atomicAdd(&DEN[(long long)d * HEADS + h], ex);
}

// agg[dst,c] += h[src,c] * (ex / max(den[dst],1e-16))
__global__ void edge_message(const float* __restrict__ EW,
                             const float* __restrict__ DEN,
                             const float* __restrict__ H,
                             const int* __restrict__ ei,
                             long long E, long long Etot,
                             float* __restrict__ AGG) {
  long long idx = (long long)blockIdx.x * blockDim.x + threadIdx.x;  // e*256+c
  if (idx >= Etot * OUTC) return;
  long long e = idx >> 8;
  int c = (int)(idx & (OUTC - 1));
  int s, d;
  edge_sd(ei, E, e, s, d);
  int h = c >> 6;
  float coeff = EW[e * HEADS + h] /
                fmaxf(DEN[(long long)d * HEADS + h], 1e-16f);
  atomicAdd(&AGG[(long long)d * OUTC + c],
            H[(long long)s * OUTC + c] * coeff);
}

// ---------------- epilogues ----------------

__global__ void bias_relu(float* __restrict__ A, const float* __restrict__ b,
                          long long n) {
  long long i = (long long)blockIdx.x * blockDim.x + threadIdx.x;
  if (i >= n) return;
  int c = (int)(i & (OUTC - 1));
  A[i] = fmaxf(A[i] + b[c], 0.f);
}

// out2[n,c] = mean over heads of agg[n,h,c] + b2[c]
__global__ void head_mean(const float* __restrict__ AGG,
                          const float* __restrict__ b2,
                          float* __restrict__ OUT2, long long n_nodes) {
  long long i = (long long)blockIdx.x * blockDim.x + threadIdx.x;  // n*64+c
  if (i >= n_nodes * HID) return;
  long long n = i >> 6;
  int c = (int)(i & 63);
  const float* p = AGG + n * OUTC + c;
  float s = p[0] + p[HID] + p[2 * HID] + p[3 * HID];
  OUT2[i] = 0.25f * s + b2[c];
}

// ---------------- pooling + MLP ----------------

__global__ void pool_accum(const float* __restrict__ OUT2,
                           const int* __restrict__ batch,
                           float* __restrict__ POOLED,
                           float* __restrict__ COUNTS, long long n_nodes) {
  long long i = (long long)blockIdx.x * blockDim.x + threadIdx.x;  // n*64+c
  if (i >= n_nodes * HID) return;
  long long n = i >> 6;
  int c = (int)(i & 63);
  int b = batch[n];
  atomicAdd(&POOLED[(long long)b * HID + c], OUT2[i]);
  if (c == 0) atomicAdd(&COUNTS[b], 1.0f);
}

__global__ void pool_divide(float* __restrict__ POOLED,
                            const float* __restrict__ COUNTS, int Bsz) {
  int i = blockIdx.x * blockDim.x + threadIdx.x;
  if (i >= Bsz * HID) return;
  POOLED[i] /= fmaxf(COUNTS[i >> 6], 1.0f);
}

__global__ void graph_mlp(const float* __restrict__ POOLED,
                          const float* __restrict__ Wm1,
                          const float* __restrict__ bm1,
                          const float* __restrict__ Wm2,
                          const float* __restrict__ bm2,
                          float* __restrict__ out) {
  __shared__ float hid[HID];
  int b = blockIdx.x, j = threadIdx.x;
  float s = 0.f;
  for (int k = 0; k < HID; ++k) s += POOLED[b * HID + k] * Wm1[k * HID + j];
  s = fmaxf(s + bm1[j], 0.f);
  hid[j] = s * Wm2[j];
  __syncthreads();
  for (int st = HID / 2; st > 0; st >>= 1) {
    if (j < st) hid[j] += hid[j + st];
    __syncthreads();
  }
  if (j == 0) out[b] = hid[0] + bm2[0];
}

// ---------------- launcher ----------------

static inline dim3 grid1d(long long n, int bs) {
  return dim3((unsigned)((n + bs - 1) / bs));
}

extern "C" void kernel_launch(void* const* d_in, const int* in_sizes, int n_in,
                              void* d_out, int out_size, void* d_ws, size_t ws_size,
                              hipStream_t stream) {
  const float* x    = (const float*)d_in[0];
  const int*   ei   = (const int*)d_in[1];
  const int*   batch= (const int*)d_in[2];
  const float* W1   = (const float*)d_in[3];
  const float* as1  = (const float*)d_in[4];
  const float* ad1  = (const float*)d_in[5];
  const float* b1   = (const float*)d_in[6];
  const float* W2   = (const float*)d_in[7];
  const float* as2  = (const float*)d_in[8];
  const float* ad2  = (const float*)d_in[9];
  const float* b2   = (const float*)d_in[10];
  const float* Wm1  = (const float*)d_in[11];
  const float* bm1  = (const float*)d_in[12];
  const float* Wm2  = (const float*)d_in[13];
  const float* bm2  = (const float*)d_in[14];
  float* out = (float*)d_out;

  const long long Nn   = in_sizes[0] / INC;   // 50000
  const long long E    = in_sizes[1] / 2;     // 800000
  const long long Etot = E + Nn;              // with self loops
  const int       Bsz  = out_size;            // 64

  // workspace layout (fp32)
  float* ws     = (float*)d_ws;
  float* H      = ws;                       // N*256  (H1, then H2, then OUT2)
  float* AGG    = H   + Nn * OUTC;          // N*256
  float* AS     = AGG + Nn * OUTC;          // N*4
  float* AD     = AS  + Nn * HEADS;         // N*4
  float* M      = AD  + Nn * HEADS;         // N*4
  float* DEN    = M   + Nn * HEADS;         // N*4
  float* EW     = DEN + Nn * HEADS;         // Etot*4
  float* POOLED = EW  + Etot * HEADS;       // B*64
  float* COUNTS = POOLED + (long long)Bsz * HID;  // B
  (void)ws_size; (void)n_in;

  const int TB = 256;
  dim3 gemmGrid((unsigned)(Nn / 16), 2);    // 8 waves/block x 2 = 16 col tiles

  // ===== layer 1 =====
  gemm_wmma_f32<INC><<<gemmGrid, TB, 0, stream>>>(x, W1, H, (int)Nn);
  attn_scores<<<grid1d(Nn * HEADS, TB), TB, 0, stream>>>(H, as1, ad1, AS, AD, Nn);
  fill_f32<<<grid1d(Nn * HEADS, TB), TB, 0, stream>>>(M, Nn * HEADS, -1e30f);
  fill_f32<<<grid1d(Nn * HEADS, TB), TB, 0, stream>>>(DEN, Nn * HEADS, 0.f);
  fill_f32<<<grid1d(Nn * OUTC, TB), TB, 0, stream>>>(AGG, Nn * OUTC, 0.f);
  edge_score_max<<<grid1d(Etot * HEADS, TB), TB, 0, stream>>>(AS, AD, ei, E, Etot, EW, M);
  edge_exp_sum  <<<grid1d(Etot * HEADS, TB), TB, 0, stream>>>(ei, E, Etot, M, EW, DEN);
  edge_message  <<<grid1d(Etot * OUTC,  TB), TB, 0, stream>>>(EW, DEN, H, ei, E, Etot, AGG);
  bias_relu<<<grid1d(Nn * OUTC, TB), TB, 0, stream>>>(AGG, b1, Nn * OUTC);

  // ===== layer 2 =====
  gemm_wmma_f32<OUTC><<<gemmGrid, TB, 0, stream>>>(AGG, W2, H, (int)Nn);
  attn_scores<<<grid1d(Nn * HEADS, TB), TB, 0, stream>>>(H, as2, ad2, AS, AD, Nn);
  fill_f32<<<grid1d(Nn * HEADS, TB), TB, 0, stream>>>(M, Nn * HEADS, -1e30f);
  fill_f32<<<grid1d(Nn * HEADS, TB), TB, 0, stream>>>(DEN, Nn * HEADS, 0.f);
  fill_f32<<<grid1d(Nn * OUTC, TB), TB, 0, stream>>>(AGG, Nn * OUTC, 0.f);
  edge_score_max<<<grid1d(Etot * HEADS, TB), TB, 0, stream>>>(AS, AD, ei, E, Etot, EW, M);
  edge_exp_sum  <<<grid1d(Etot * HEADS, TB), TB, 0, stream>>>(ei, E, Etot, M, EW, DEN);
  edge_message  <<<grid1d(Etot * OUTC,  TB), TB, 0, stream>>>(EW, DEN, H, ei, E, Etot, AGG);

  // head mean + bias -> OUT2 (reuse H region)
  float* OUT2 = H;
  head_mean<<<grid1d(Nn * HID, TB), TB, 0, stream>>>(AGG, b2, OUT2, Nn);

  // ===== pooling + graph MLP =====
  fill_f32<<<grid1d((long long)Bsz * HID, TB), TB, 0, stream>>>(POOLED, (long long)Bsz * HID, 0.f);
  fill_f32<<<grid1d(Bsz, 64), 64, 0, stream>>>(COUNTS, Bsz, 0.f);
  pool_accum<<<grid1d(Nn * HID, TB), TB, 0, stream>>>(OUT2, batch, POOLED, COUNTS, Nn);
  pool_divide<<<grid1d((long long)Bsz * HID, TB), TB, 0, stream>>>(POOLED, COUNTS, Bsz);
  graph_mlp<<<Bsz, HID, 0, stream>>>(POOLED, Wm1, bm1, Wm2, bm2, out);
}